// NCC_Loss_8675833938240
// MI455X (gfx1250) — compile-verified
//
#include <hip/hip_runtime.h>
#include <stdint.h>

// ---------------------------------------------------------------------------
// NCC loss, fused single-pass tiled implementation for gfx1250 (MI455X).
//   traffic: ~145MB (128MB + 13% halo)  ->  ~6.2us @ 23.3 TB/s (HBM bound)
//   compute: separable 9x9 box sums via sliding windows (negligible vs BW)
// CDNA5 features: TDM tensor_load_to_lds (async tile DMA into LDS, interior
// tiles), s_wait_tensorcnt, v_wmma_f32_16x16x4_f32 (block reduction).
// ---------------------------------------------------------------------------

#define TS    64          // output tile side
#define HALO  4
#define TW    72          // TS + 2*HALO, LDS tile stride (fixed by TDM layout)
#define HST   65          // padded stride of H buffer (bank-conflict free)
#define IMG   1024
#define NBATCH 16
#define NPIX  (16.0f * 1024.0f * 1024.0f)
#define WIN_INV (1.0f / 81.0f)
#define NCC_EPS 1e-5f

typedef float v2f __attribute__((ext_vector_type(2)));
typedef float v8f __attribute__((ext_vector_type(8)));
typedef unsigned int v4u __attribute__((ext_vector_type(4)));
typedef int v8i __attribute__((ext_vector_type(8)));
typedef int v4i __attribute__((ext_vector_type(4)));

// low 32 bits of a flat LDS address == LDS byte offset (aperture in [63:32])
__device__ inline unsigned lds_off(const void* p) {
  return (unsigned)(uintptr_t)p;
}

// ---- Tensor Data Mover: DMA a 72x72 f32 tile (row stride 1024 elems) from
// global memory into LDS at lds_byte_off, contiguous 72-elem rows.
// Descriptor per CDNA5 ISA (groups 0..3).
// This toolchain's builtin takes 6 args (clang-23 / therock form):
//   (v4u g0, v8i g1, v4i g2, v4i g3, v8i g4, i32 cpol)
__device__ inline void tdm_load_tile_f32(const float* gsrc, unsigned lds_byte_off,
                                         unsigned remW, unsigned remH) {
  unsigned long long ga = (unsigned long long)(uintptr_t)gsrc;
  v4u g0;
  g0[0] = 1u;                                               // count=1 (valid), user mode
  g0[1] = lds_byte_off;                                     // lds_addr (bytes)
  g0[2] = (unsigned)(ga & 0xFFFFFFFFull);                   // global_addr[31:0]
  g0[3] = (unsigned)((ga >> 32) & 0x1FFFFFFull) | (2u << 30); // addr[56:32] | type=2
  v8i g1;
  g1[0] = (int)(2u << 16);                                  // data_size=4B; no iterate/pad
  g1[1] = (int)((remW & 0xFFFFu) << 16);                    // tensor_dim0 lo16 (bits 63:48)
  g1[2] = (int)((remW >> 16) | ((remH & 0xFFFFu) << 16));   // dim0 hi16 | dim1 lo16
  g1[3] = (int)((remH >> 16) | ((unsigned)TW << 16));       // dim1 hi16 | tile_dim0=72
  g1[4] = (int)TW;                                          // tile_dim1=72, tile_dim2=0
  g1[5] = (int)IMG;                                         // tensor_dim0_stride lo32 = 1024
  g1[6] = 0;                                                // stride0 hi16 | stride1 lo16
  g1[7] = (int)((IMG * IMG) >> 16);                         // tensor_dim1_stride hi32 (1M)
  v4i g2; g2[0] = 1; g2[1] = 0; g2[2] = 0; g2[3] = 0;       // tensor_dim2=1 (unused)
  v4i g3; g3[0] = 0; g3[1] = 0; g3[2] = 0; g3[3] = 0;
  v8i g4; g4[0] = 0; g4[1] = 0; g4[2] = 0; g4[3] = 0;
          g4[4] = 0; g4[5] = 0; g4[6] = 0; g4[7] = 0;
  __builtin_amdgcn_tensor_load_to_lds(g0, g1, g2, g3, g4, 0);
}

// ---- Block-wide sum via V_WMMA_F32_16X16X4_F32 (layout-robust):
// D = A x ones + 0  ->  D[m][n] = rowsum_A[m] for every n.
// A vgpr0 = acc, vgpr1 = 0: rowsum[m] = acc[lane m] + acc[lane m+16].
// D (documented C/D layout): lane 0 vgprs 0..7 = rowsum[0..7],
//                            lane 16 vgprs 0..7 = rowsum[8..15].
// So (sum of 8 D vgprs) at lanes 0 and 16 partitions the wave sum exactly.
// Requires EXEC all ones -> call unconditionally with full blocks.
__device__ inline float block_reduce_sum(float v, float* sRed) {
  v2f a; a[0] = v;    a[1] = 0.0f;
  v2f o; o[0] = 1.0f; o[1] = 1.0f;
  v8f c = {};
  v8f d = __builtin_amdgcn_wmma_f32_16x16x4_f32(false, a, false, o, (short)0, c,
                                                false, false);
  float s8 = ((d[0] + d[1]) + (d[2] + d[3])) + ((d[4] + d[5]) + (d[6] + d[7]));
  const int tid  = threadIdx.x;
  const int lane = tid & 31;
  if ((lane & 15) == 0)                    // lanes 0 and 16 of each wave
    sRed[((tid >> 5) << 1) + (lane >> 4)] = s8;
  __syncthreads();
  float tot = 0.0f;
  if (tid == 0) {
#pragma unroll
    for (int i = 0; i < 16; ++i) tot += sRed[i];   // 8 waves * 2 partials
  }
  return tot;
}

// channel value: 0=I, 1=J, 2=I*J, 3=I*I, 4=J*J
template <int C>
__device__ inline float qv(const float* __restrict__ sI, const float* __restrict__ sJ,
                           int idx) {
  if constexpr (C == 0) return sI[idx];
  else if constexpr (C == 1) return sJ[idx];
  else if constexpr (C == 2) return sI[idx] * sJ[idx];
  else if constexpr (C == 3) { float v = sI[idx]; return v * v; }
  else                       { float v = sJ[idx]; return v * v; }
}

// Pass A: horizontal 9-tap sliding sums. H[row][x] = sum_{t=x..x+8} q[row][t],
// row = 0..71 (incl. halo rows), x = 0..63. 288 strips of 16 outputs.
template <int C>
__device__ inline void passA(const float* __restrict__ sI, const float* __restrict__ sJ,
                             float* __restrict__ sH, int tid) {
  for (int s = tid; s < TW * (TS / 16); s += 256) {
    const int row  = s >> 2;
    const int x0   = (s & 3) << 4;
    const int base = row * TW + x0;
    float acc = 0.0f;
#pragma unroll
    for (int d = 0; d < 9; ++d) acc += qv<C>(sI, sJ, base + d);
    sH[row * HST + x0] = acc;
#pragma unroll
    for (int k = 1; k < 16; ++k) {
      acc += qv<C>(sI, sJ, base + 8 + k) - qv<C>(sI, sJ, base + k - 1);
      sH[row * HST + x0 + k] = acc;
    }
  }
}

// Pass B: vertical 9-tap sliding sums down one column; 16 outputs per thread.
__device__ inline void passB(const float* __restrict__ sH, int x, int Y0,
                             float out[16]) {
  float s = 0.0f;
#pragma unroll
  for (int d = 0; d < 9; ++d) s += sH[(Y0 + d) * HST + x];
  out[0] = s;
#pragma unroll
  for (int k = 1; k < 16; ++k) {
    s += sH[(Y0 + 8 + k) * HST + x] - sH[(Y0 + k - 1) * HST + x];
    out[k] = s;
  }
}

__global__ __launch_bounds__(256) void ncc_tile_kernel(const float* __restrict__ pred,
                                                       const float* __restrict__ targ,
                                                       float* __restrict__ partial) {
  __shared__ float sI[TW * TW];     // 20736 B
  __shared__ float sJ[TW * TW];     // 20736 B
  __shared__ float sH[TW * HST];    // 18720 B (one channel at a time)
  __shared__ float sRed[16];

  const int tid   = threadIdx.x;
  const int b     = blockIdx.x;
  const int batch = b >> 8;
  const int tile  = b & 255;
  const int ty = tile >> 4, tx = tile & 15;
  const int gx0 = tx * TS - HALO;   // tile origin incl. halo (may be < 0)
  const int gy0 = ty * TS - HALO;
  const float* I = pred + (size_t)batch * (IMG * IMG);
  const float* J = targ + (size_t)batch * (IMG * IMG);

  const bool interior = (tx >= 1) & (tx <= 14) & (ty >= 1) & (ty <= 14);
  if (interior) {
    // whole 72x72 halo tile is in-bounds: async DMA it with the TDM (wave 0)
    if (tid < 32) {
      const float* gI = I + (size_t)gy0 * IMG + gx0;
      const float* gJ = J + (size_t)gy0 * IMG + gx0;
      tdm_load_tile_f32(gI, lds_off(sI), (unsigned)(IMG - gx0), (unsigned)(IMG - gy0));
      tdm_load_tile_f32(gJ, lds_off(sJ), (unsigned)(IMG - gx0), (unsigned)(IMG - gy0));
      __builtin_amdgcn_s_wait_tensorcnt(0);
    }
  } else {
    // border tile: guarded VALU loads implement the zero padding
    for (int i = tid; i < TW * TW; i += 256) {
      const int r  = i / TW;
      const int cc = i - r * TW;
      const int gy = gy0 + r, gx = gx0 + cc;
      float vi = 0.0f, vj = 0.0f;
      if (gy >= 0 && gy < IMG && gx >= 0 && gx < IMG) {
        const size_t o = (size_t)gy * IMG + gx;
        vi = I[o];
        vj = J[o];
      }
      sI[i] = vi;
      sJ[i] = vj;
    }
  }
  __syncthreads();

  const int x  = tid & 63;          // output column (lanes: distinct banks)
  const int Y0 = (tid >> 6) << 4;   // first of 16 output rows

  float SIv[16], SJv[16], CRv[16], IVv[16], T[16];

  // channel 0: S_I
  passA<0>(sI, sJ, sH, tid);
  __syncthreads();
  passB(sH, x, Y0, SIv);
  __syncthreads();

  // channel 1: S_J
  passA<1>(sI, sJ, sH, tid);
  __syncthreads();
  passB(sH, x, Y0, SJv);
  __syncthreads();

  // channel 2: S_IJ -> cross = S_IJ - S_I*S_J/81
  passA<2>(sI, sJ, sH, tid);
  __syncthreads();
  passB(sH, x, Y0, T);
#pragma unroll
  for (int k = 0; k < 16; ++k) CRv[k] = T[k] - SIv[k] * SJv[k] * WIN_INV;
  __syncthreads();

  // channel 3: S_II -> I_var = S_II - S_I^2/81
  passA<3>(sI, sJ, sH, tid);
  __syncthreads();
  passB(sH, x, Y0, T);
#pragma unroll
  for (int k = 0; k < 16; ++k) IVv[k] = T[k] - SIv[k] * SIv[k] * WIN_INV;
  __syncthreads();

  // channel 4: S_JJ -> J_var, then cc = cross^2 / (I_var*J_var + eps)
  passA<4>(sI, sJ, sH, tid);
  __syncthreads();
  passB(sH, x, Y0, T);
  float acc = 0.0f;
#pragma unroll
  for (int k = 0; k < 16; ++k) {
    const float jv = T[k] - SJv[k] * SJv[k] * WIN_INV;
    const float cr = CRv[k];
    acc += (cr * cr) / (IVv[k] * jv + NCC_EPS);
  }

  const float tot = block_reduce_sum(acc, sRed);
  if (tid == 0) partial[b] = tot;
}

__global__ __launch_bounds__(256) void ncc_finalize_kernel(const float* __restrict__ partial,
                                                           float* __restrict__ out,
                                                           int n) {
  __shared__ float sRed[16];
  float v = 0.0f;
  for (int i = threadIdx.x; i < n; i += 256) v += partial[i];
  const float tot = block_reduce_sum(v, sRed);
  if (threadIdx.x == 0) out[0] = -tot / NPIX;
}

extern "C" void kernel_launch(void* const* d_in, const int* in_sizes, int n_in,
                              void* d_out, int out_size, void* d_ws, size_t ws_size,
                              hipStream_t stream) {
  (void)in_sizes; (void)n_in; (void)out_size; (void)ws_size;
  const float* pred = (const float*)d_in[0];
  const float* targ = (const float*)d_in[1];
  float* out     = (float*)d_out;
  float* partial = (float*)d_ws;            // 4096 f32 partial sums

  const int nblocks = NBATCH * (IMG / TS) * (IMG / TS);   // 16 * 16 * 16 = 4096
  ncc_tile_kernel<<<nblocks, 256, 0, stream>>>(pred, targ, partial);
  ncc_finalize_kernel<<<1, 256, 0, stream>>>(partial, out, nblocks);
}